// SoftMorphTransform3D_16956530885077
// MI455X (gfx1250) — compile-verified
//
#include <hip/hip_runtime.h>

// Problem geometry (from reference: img = (2,1,128,256,256) fp32)
#define NB 2
#define DD 128
#define HH 256
#define WW 256

// Tile geometry: fused 4-step stencil (radius 1 each) -> halo 4
#define TD 8
#define TH 16
#define TW 32
#define RHALO 4
#define BD (TD + 2*RHALO)    // 16
#define BH (TH + 2*RHALO)    // 24
#define BW (TW + 2*RHALO)    // 40 floats = 160B rows -> every 4-float chunk 16B aligned
#define BUF_ELEMS (BD*BH*BW) // 15360 floats = 60 KB per buffer
#define NTHREADS 256

#define BIDX(z,y,x) (((z)*BH + (y))*BW + (x))

// Generic pointer -> raw LDS byte offset (VDST operand of async-to-LDS ops).
__device__ __forceinline__ unsigned lds_offset(const void* p) {
    return (unsigned)(unsigned long long)
        (__attribute__((address_space(3))) const char*)p;
}

// Erode (x-vectorized by 4): out = c*c * prod(6 face neighbors).
// Out-of-volume cells get padNext (the pad constant the NEXT step expects).
template<int M>
__device__ __forceinline__ void erode_step(const float* __restrict__ a,
                                           float* __restrict__ b,
                                           int tz0, int ty0, int tx0,
                                           int tid, float padNext) {
    constexpr int RZ = BD - 2*M, RY = BH - 2*M;
    constexpr int XC = BW / 4;          // full-width x chunks; edge lanes are
    constexpr int N  = RZ * RY * XC;    // garbage but provably never read.
    for (int i = tid; i < N; i += NTHREADS) {
        int xc = i % XC, t = i / XC;
        int y  = t % RY + M, z = t / RY + M;
        int x4 = xc * 4;
        int base = BIDX(z, y, x4);
        float4 c  = *(const float4*)(a + base);
        float4 zm = *(const float4*)(a + base - BH*BW);
        float4 zp = *(const float4*)(a + base + BH*BW);
        float4 ym = *(const float4*)(a + base - BW);
        float4 yp = *(const float4*)(a + base + BW);
        float  xm = a[(x4 == 0)      ? base     : base - 1];  // clamped: feeds garbage lane only
        float  xp = a[(x4 + 4 >= BW) ? base + 3 : base + 4];
        float4 v;
        v.x = c.x*c.x * zm.x*zp.x * ym.x*yp.x * xm  * c.y;
        v.y = c.y*c.y * zm.y*zp.y * ym.y*yp.y * c.x * c.z;
        v.z = c.z*c.z * zm.z*zp.z * ym.z*yp.z * c.y * c.w;
        v.w = c.w*c.w * zm.w*zp.w * ym.w*yp.w * c.z * xp;
        int gz = tz0 + z, gy = ty0 + y, gx = tx0 + x4;
        bool zyin = ((unsigned)gz < DD) & ((unsigned)gy < HH);
        v.x = (zyin && (unsigned)(gx + 0) < WW) ? v.x : padNext;
        v.y = (zyin && (unsigned)(gx + 1) < WW) ? v.y : padNext;
        v.z = (zyin && (unsigned)(gx + 2) < WW) ? v.z : padNext;
        v.w = (zyin && (unsigned)(gx + 3) < WW) ? v.w : padNext;
        *(float4*)(b + base) = v;
    }
}

// Dilate (x-vectorized by 4): out = 1 - (1-c) * prod(1 - neighbor).
template<int M>
__device__ __forceinline__ void dilate_step(const float* __restrict__ a,
                                            float* __restrict__ b,
                                            int tz0, int ty0, int tx0,
                                            int tid, float padNext) {
    constexpr int RZ = BD - 2*M, RY = BH - 2*M;
    constexpr int XC = BW / 4;
    constexpr int N  = RZ * RY * XC;
    for (int i = tid; i < N; i += NTHREADS) {
        int xc = i % XC, t = i / XC;
        int y  = t % RY + M, z = t / RY + M;
        int x4 = xc * 4;
        int base = BIDX(z, y, x4);
        float4 c  = *(const float4*)(a + base);
        float4 zm = *(const float4*)(a + base - BH*BW);
        float4 zp = *(const float4*)(a + base + BH*BW);
        float4 ym = *(const float4*)(a + base - BW);
        float4 yp = *(const float4*)(a + base + BW);
        float  xm = a[(x4 == 0)      ? base     : base - 1];
        float  xp = a[(x4 + 4 >= BW) ? base + 3 : base + 4];
        float4 v;
        v.x = 1.0f - (1.0f-c.x)*(1.0f-zm.x)*(1.0f-zp.x)*(1.0f-ym.x)*(1.0f-yp.x)*(1.0f-xm) *(1.0f-c.y);
        v.y = 1.0f - (1.0f-c.y)*(1.0f-zm.y)*(1.0f-zp.y)*(1.0f-ym.y)*(1.0f-yp.y)*(1.0f-c.x)*(1.0f-c.z);
        v.z = 1.0f - (1.0f-c.z)*(1.0f-zm.z)*(1.0f-zp.z)*(1.0f-ym.z)*(1.0f-yp.z)*(1.0f-c.y)*(1.0f-c.w);
        v.w = 1.0f - (1.0f-c.w)*(1.0f-zm.w)*(1.0f-zp.w)*(1.0f-ym.w)*(1.0f-yp.w)*(1.0f-c.z)*(1.0f-xp);
        int gz = tz0 + z, gy = ty0 + y, gx = tx0 + x4;
        bool zyin = ((unsigned)gz < DD) & ((unsigned)gy < HH);
        v.x = (zyin && (unsigned)(gx + 0) < WW) ? v.x : padNext;
        v.y = (zyin && (unsigned)(gx + 1) < WW) ? v.y : padNext;
        v.z = (zyin && (unsigned)(gx + 2) < WW) ? v.z : padNext;
        v.w = (zyin && (unsigned)(gx + 3) < WW) ? v.w : padNext;
        *(float4*)(b + base) = v;
    }
}

// Final dilate: computes only the in-volume tile, streams b128 stores to global.
__device__ __forceinline__ void dilate_final(const float* __restrict__ a,
                                             float* __restrict__ dst,
                                             int tz0, int ty0, int tx0,
                                             int tid) {
    constexpr int XC = TW / 4;            // 8
    constexpr int N  = TD * TH * XC;      // 1024
    for (int i = tid; i < N; i += NTHREADS) {
        int xc = i % XC, t = i / XC;
        int y  = t % TH + RHALO, z = t / TH + RHALO;
        int x4 = RHALO + xc * 4;          // in [4,36): x-1 and x+4 stay in-row
        int base = BIDX(z, y, x4);
        float4 c  = *(const float4*)(a + base);
        float4 zm = *(const float4*)(a + base - BH*BW);
        float4 zp = *(const float4*)(a + base + BH*BW);
        float4 ym = *(const float4*)(a + base - BW);
        float4 yp = *(const float4*)(a + base + BW);
        float  xm = a[base - 1];
        float  xp = a[base + 4];
        float4 v;
        v.x = 1.0f - (1.0f-c.x)*(1.0f-zm.x)*(1.0f-zp.x)*(1.0f-ym.x)*(1.0f-yp.x)*(1.0f-xm) *(1.0f-c.y);
        v.y = 1.0f - (1.0f-c.y)*(1.0f-zm.y)*(1.0f-zp.y)*(1.0f-ym.y)*(1.0f-yp.y)*(1.0f-c.x)*(1.0f-c.z);
        v.z = 1.0f - (1.0f-c.z)*(1.0f-zm.z)*(1.0f-zp.z)*(1.0f-ym.z)*(1.0f-yp.z)*(1.0f-c.y)*(1.0f-c.w);
        v.w = 1.0f - (1.0f-c.w)*(1.0f-zm.w)*(1.0f-zp.w)*(1.0f-ym.w)*(1.0f-yp.w)*(1.0f-c.z)*(1.0f-xp);
        int gz = tz0 + z, gy = ty0 + y, gx = tx0 + x4;  // always in volume, 16B aligned
        *(float4*)(dst + ((size_t)gz * HH + gy) * WW + gx) = v;
    }
}

__global__ __launch_bounds__(NTHREADS)
void soft_morph_open3d(const float* __restrict__ img, float* __restrict__ out) {
    __shared__ __align__(16) float buf0[BUF_ELEMS];
    __shared__ __align__(16) float buf1[BUF_ELEMS];

    const int tid = threadIdx.x;
    const int ZT  = DD / TD;
    const int bat = blockIdx.z / ZT;
    const int tz0 = (blockIdx.z % ZT) * TD - RHALO;
    const int ty0 = blockIdx.y * TH - RHALO;
    const int tx0 = blockIdx.x * TW - RHALO;

    const float* src = img + (size_t)bat * DD * HH * WW;
    float*       dst = out + (size_t)bat * DD * HH * WW;

    // ---- Stage input tile (with 4-deep halo) into LDS via async-to-LDS ----
    // 16B chunks; x-chunks are 4-aligned and W%4==0, so a chunk never
    // straddles the volume x-boundary. Out-of-volume cells get erode pad 1.0.
    constexpr int XCH    = BW / 4;           // 10 chunks per row
    constexpr int CHUNKS = BD * BH * XCH;    // 3840
    for (int c = tid; c < CHUNKS; c += NTHREADS) {
        int xc = c % XCH, t = c / XCH;
        int y  = t % BH,  z = t / BH;
        int gz = tz0 + z, gy = ty0 + y, gx = tx0 + xc * 4;
        float* lp = &buf0[BIDX(z, y, xc * 4)];
        if ((unsigned)gz < DD && (unsigned)gy < HH && (unsigned)gx < WW) {
            const float* gp = src + ((size_t)gz * HH + gy) * WW + gx;
            unsigned loff = lds_offset(lp);
            asm volatile("global_load_async_to_lds_b128 %0, %1, off"
                         :: "v"(loff), "v"(gp) : "memory");
        } else {
            lp[0] = 1.0f; lp[1] = 1.0f; lp[2] = 1.0f; lp[3] = 1.0f;
        }
    }
    asm volatile("s_wait_asynccnt 0x0" ::: "memory");
    __syncthreads();

    // ---- Fused 4-step pipeline through LDS ping-pong buffers ----
    erode_step<1>(buf0, buf1, tz0, ty0, tx0, tid, 1.0f);  // next step erodes: pad 1
    __syncthreads();
    erode_step<2>(buf1, buf0, tz0, ty0, tx0, tid, 0.0f);  // next step dilates: pad 0
    __syncthreads();
    dilate_step<3>(buf0, buf1, tz0, ty0, tx0, tid, 0.0f); // next step dilates: pad 0
    __syncthreads();
    dilate_final(buf1, dst, tz0, ty0, tx0, tid);
}

extern "C" void kernel_launch(void* const* d_in, const int* in_sizes, int n_in,
                              void* d_out, int out_size, void* d_ws, size_t ws_size,
                              hipStream_t stream) {
    (void)in_sizes; (void)n_in; (void)out_size; (void)d_ws; (void)ws_size;
    const float* img = (const float*)d_in[0];
    float*       out = (float*)d_out;
    dim3 grid(WW / TW, HH / TH, (DD / TD) * NB);  // (8, 16, 32)
    soft_morph_open3d<<<grid, NTHREADS, 0, stream>>>(img, out);
}